// HardBatchMiningTripletLoss_66108136620072
// MI455X (gfx1250) — compile-verified
//
#include <hip/hip_runtime.h>
#include <math.h>

typedef __attribute__((ext_vector_type(2))) float v2f;
typedef __attribute__((ext_vector_type(4))) float v4f;
typedef __attribute__((ext_vector_type(8))) float v8f;

#define BATCH 8192
#define DIM   256
#define LDA   260            // padded A strip: 260 % 64 == 4 -> conflict-free b64 reads
#define NTILES (BATCH / 16)  // 512
#define MROWS 32             // rows per block (two 16-row M-tiles)
#define WAVES_PER_BLOCK 8
#define NLABELS 512
#define MARGIN 0.3f

// ---------------------------------------------------------------------------
// K1a: sq[i] = sum_j X[i,j]^2   (one wave32 per row)
// ---------------------------------------------------------------------------
__global__ void row_sqnorm_kernel(const float* __restrict__ X, float* __restrict__ sq) {
    const int wave = threadIdx.x >> 5;
    const int lane = threadIdx.x & 31;
    const int row  = blockIdx.x * WAVES_PER_BLOCK + wave;
    const float* p = X + (size_t)row * DIM;
    float s = 0.0f;
#pragma unroll
    for (int j = 0; j < DIM / 32; ++j) {
        float v = p[lane + 32 * j];
        s = fmaf(v, v, s);
    }
#pragma unroll
    for (int off = 16; off >= 1; off >>= 1)
        s += __shfl_xor(s, off, 32);
    if (lane == 0) sq[row] = s;
}

// ---------------------------------------------------------------------------
// K1b: valid-label histogram. pos_count[i] = hist[t_i], neg_count[i] =
// nvalid - pos_count[i] — counts depend only on labels, so they never need to
// ride through the GEMM. Integer atomics: order-independent => deterministic.
// hist[NLABELS] holds nvalid.
// ---------------------------------------------------------------------------
__global__ void label_count_kernel(const int* __restrict__ tgt, int* __restrict__ hist) {
    const int tid = threadIdx.x;
    for (int i = tid; i <= NLABELS; i += 256) hist[i] = 0;
    __threadfence();
    __syncthreads();
    for (int i = tid; i < BATCH; i += 256) {
        int t = tgt[i];
        if (t != -1) {
            atomicAdd(&hist[t], 1);
            atomicAdd(&hist[NLABELS], 1);
        }
    }
}

// masked min/max update for one C tile (8 rows) in e-space, e = sq_n - 2*g
__device__ __forceinline__ void upd8(const v8f& c, float sqn, int tn_, bool valid,
                                     const int (&tm_)[8],
                                     float (&pmin)[8], float (&pmax)[8],
                                     float (&nmax)[8], float (&nnm)[8]) {
    const float INF = __builtin_inff();
#pragma unroll
    for (int r = 0; r < 8; ++r) {
        float e = fmaf(-2.0f, c[r], sqn);
        bool same = (tm_[r] == tn_);
        bool pos  = same && valid;
        bool neg  = (!same) && valid;
        pmin[r] = fminf(pmin[r], pos ? e : INF);
        pmax[r] = fmaxf(pmax[r], pos ? e : -INF);
        nmax[r] = fmaxf(nmax[r], neg ? e : -INF);
        nnm[r]  = fminf(nnm[r],  neg ? INF : e);
    }
}

// ---------------------------------------------------------------------------
// K2: fused Gram (fp32 WMMA 16x16x4) + hard mining.
// 32-row M strip per block (B traffic halved vs 16-row), 2 N-tiles per wave
// iteration => 2x2 = 4 independent WMMA chains; A shared from LDS.
// ---------------------------------------------------------------------------
__global__ void __launch_bounds__(256)
mine_kernel(const float* __restrict__ X, const int* __restrict__ tgt,
            const float* __restrict__ sq, const int* __restrict__ hist,
            float* __restrict__ rowloss) {
    __shared__ float ldsA[MROWS * LDA];
    __shared__ float red[WAVES_PER_BLOCK][MROWS][4];

    const int tid  = threadIdx.x;
    const int wave = tid >> 5;
    const int lane = tid & 31;
    const int row0 = blockIdx.x * MROWS;

    // Stage the 32x256 A strip into LDS (float4, padded rows).
    for (int i = tid; i < MROWS * (DIM / 4); i += 256) {
        int r  = i >> 6;
        int c4 = i & 63;
        v4f v = *(const v4f*)&X[(size_t)(row0 + r) * DIM + c4 * 4];
        *(v4f*)&ldsA[r * LDA + c4 * 4] = v;
    }
    __syncthreads();

    const int nIdx = lane & 15;
    const int hi   = lane >> 4;
    const int kofs = 2 * hi;

    // Row labels for this lane's C rows: m = 16*t + r + 8*hi.
    int tm0[8], tm1[8];
#pragma unroll
    for (int r = 0; r < 8; ++r) {
        tm0[r] = tgt[row0 + r + 8 * hi];
        tm1[r] = tgt[row0 + 16 + r + 8 * hi];
    }

    const float INF = __builtin_inff();
    float pmin0[8], pmax0[8], nmax0[8], nnm0[8];
    float pmin1[8], pmax1[8], nmax1[8], nnm1[8];
#pragma unroll
    for (int r = 0; r < 8; ++r) {
        pmin0[r] = INF;  pmax0[r] = -INF; nmax0[r] = -INF; nnm0[r] = INF;
        pmin1[r] = INF;  pmax1[r] = -INF; nmax1[r] = -INF; nnm1[r] = INF;
    }

    for (int tn = 2 * wave; tn < NTILES; tn += 2 * WAVES_PER_BLOCK) {
        const int colA = tn * 16;
        const int colB = colA + 16;
        const float* browA = X + (size_t)(colA + nIdx) * DIM;  // B = X^T tiles
        const float* browB = browA + (size_t)16 * DIM;
        const int   tnA = tgt[colA + nIdx];
        const int   tnB = tgt[colB + nIdx];
        const float sqA = sq[colA + nIdx];
        const float sqB = sq[colB + nIdx];
        const bool  vA  = (tnA != -1);
        const bool  vB  = (tnB != -1);

        if (tn + 2 * WAVES_PER_BLOCK < NTILES) {
            __builtin_prefetch(browA + (size_t)16 * 2 * WAVES_PER_BLOCK * DIM, 0, 1);
        }

        v8f c00 = {}, c01 = {}, c10 = {}, c11 = {};   // c[mTile][nTile]
#pragma unroll 4
        for (int s = 0; s < DIM / 4; ++s) {
            v2f a0 = *(const v2f*)&ldsA[nIdx * LDA + 4 * s + kofs];
            v2f a1 = *(const v2f*)&ldsA[(nIdx + 16) * LDA + 4 * s + kofs];
            v2f b0 = *(const v2f*)&browA[4 * s + kofs];
            v2f b1 = *(const v2f*)&browB[4 * s + kofs];
            c00 = __builtin_amdgcn_wmma_f32_16x16x4_f32(false, a0, false, b0,
                                                        (short)0, c00, false, false);
            c10 = __builtin_amdgcn_wmma_f32_16x16x4_f32(false, a1, false, b0,
                                                        (short)0, c10, false, false);
            c01 = __builtin_amdgcn_wmma_f32_16x16x4_f32(false, a0, false, b1,
                                                        (short)0, c01, false, false);
            c11 = __builtin_amdgcn_wmma_f32_16x16x4_f32(false, a1, false, b1,
                                                        (short)0, c11, false, false);
        }

        upd8(c00, sqA, tnA, vA, tm0, pmin0, pmax0, nmax0, nnm0);
        upd8(c01, sqB, tnB, vB, tm0, pmin0, pmax0, nmax0, nnm0);
        upd8(c10, sqA, tnA, vA, tm1, pmin1, pmax1, nmax1, nnm1);
        upd8(c11, sqB, tnB, vB, tm1, pmin1, pmax1, nmax1, nnm1);
    }

    // Reduce over the 16 n-lanes (xor stays inside each 16-lane half).
#pragma unroll
    for (int off = 1; off < 16; off <<= 1) {
#pragma unroll
        for (int r = 0; r < 8; ++r) {
            pmin0[r] = fminf(pmin0[r], __shfl_xor(pmin0[r], off, 32));
            pmax0[r] = fmaxf(pmax0[r], __shfl_xor(pmax0[r], off, 32));
            nmax0[r] = fmaxf(nmax0[r], __shfl_xor(nmax0[r], off, 32));
            nnm0[r]  = fminf(nnm0[r],  __shfl_xor(nnm0[r],  off, 32));
            pmin1[r] = fminf(pmin1[r], __shfl_xor(pmin1[r], off, 32));
            pmax1[r] = fmaxf(pmax1[r], __shfl_xor(pmax1[r], off, 32));
            nmax1[r] = fmaxf(nmax1[r], __shfl_xor(nmax1[r], off, 32));
            nnm1[r]  = fminf(nnm1[r],  __shfl_xor(nnm1[r],  off, 32));
        }
    }

    if (nIdx == 0) {   // lanes 0 and 16
#pragma unroll
        for (int r = 0; r < 8; ++r) {
            int m0 = r + 8 * hi;
            red[wave][m0][0]      = pmin0[r];
            red[wave][m0][1]      = pmax0[r];
            red[wave][m0][2]      = nmax0[r];
            red[wave][m0][3]      = nnm0[r];
            red[wave][m0 + 16][0] = pmin1[r];
            red[wave][m0 + 16][1] = pmax1[r];
            red[wave][m0 + 16][2] = nmax1[r];
            red[wave][m0 + 16][3] = nnm1[r];
        }
    }
    __syncthreads();

    if (tid < MROWS) {
        float a0 = INF, a1 = -INF, a2 = -INF, a3 = INF;
#pragma unroll
        for (int w = 0; w < WAVES_PER_BLOCK; ++w) {
            a0 = fminf(a0, red[w][tid][0]);
            a1 = fmaxf(a1, red[w][tid][1]);
            a2 = fmaxf(a2, red[w][tid][2]);
            a3 = fminf(a3, red[w][tid][3]);
        }
        const int row = row0 + tid;
        const int t   = tgt[row];
        const int pc  = (t == -1) ? 0 : hist[t];     // pos_count (incl. diagonal)
        const int nc  = hist[NLABELS] - pc;          // neg_count = nvalid - pos_count
        float e_pos = (pc > 1) ? a0 : a1;            // pos_count > 1 ? pos_min : pos_max
        float e_neg = (nc > 0) ? a2 : a3;            // neg_count > 0 ? neg_max : notneg_min
        const float sqm = sq[row];
        // dist(x) = sqrt(max(sqrt(max(x,1e-24)),1e-12)) applied post-selection
        float dpos = sqrtf(fmaxf(sqrtf(fmaxf(sqm + e_pos, 1e-24f)), 1e-12f));
        float dneg = sqrtf(fmaxf(sqrtf(fmaxf(sqm + e_neg, 1e-24f)), 1e-12f));
        rowloss[row] = fmaxf(dpos - dneg + MARGIN, 0.0f);
    }
}

// ---------------------------------------------------------------------------
// K3: deterministic mean over the 8192 per-row losses (fixed tree order).
// ---------------------------------------------------------------------------
__global__ void finalize_kernel(const float* __restrict__ rowloss, float* __restrict__ out) {
    __shared__ float sm[256];
    const int tid = threadIdx.x;
    float s = 0.0f;
    for (int i = tid; i < BATCH; i += 256) s += rowloss[i];
    sm[tid] = s;
    __syncthreads();
    for (int off = 128; off >= 1; off >>= 1) {
        if (tid < off) sm[tid] += sm[tid + off];
        __syncthreads();
    }
    if (tid == 0) out[0] = sm[0] / (float)BATCH;
}

extern "C" void kernel_launch(void* const* d_in, const int* in_sizes, int n_in,
                              void* d_out, int out_size, void* d_ws, size_t ws_size,
                              hipStream_t stream) {
    const float* X   = (const float*)d_in[0];  // [8192, 256] f32
    const int*   tgt = (const int*)d_in[1];    // [8192] int
    float* sq      = (float*)d_ws;             // [8192] f32
    float* rowloss = sq + BATCH;               // [8192] f32
    int*   hist    = (int*)(rowloss + BATCH);  // [513]  int
    float* out     = (float*)d_out;            // scalar f32

    row_sqnorm_kernel<<<BATCH / WAVES_PER_BLOCK, 256, 0, stream>>>(X, sq);
    label_count_kernel<<<1, 256, 0, stream>>>(tgt, hist);
    mine_kernel<<<BATCH / MROWS, 256, 0, stream>>>(X, tgt, sq, hist, rowloss);
    finalize_kernel<<<1, 256, 0, stream>>>(rowloss, out);
}